// BatchWiseTripletDistanceLoss_29231547417152
// MI455X (gfx1250) — compile-verified
//
#include <hip/hip_runtime.h>
#include <math.h>

typedef __attribute__((ext_vector_type(2))) float v2f;
typedef __attribute__((ext_vector_type(8))) float v8f;
typedef int av4i __attribute__((vector_size(16)));   // GCC-style int4, matches builtin param

#define N_ROWS   4096
#define DIM      1024
#define NCLASS   512
#define N_NEGS   3679          // int(0.9 * (4096 - 8))
#define MARGIN   0.15f
#define LSTRIDE  36            // LDS panel stride in floats (even, 16B-aligned rows)
#define NCHUNK   (DIM / 32)    // 32 K-chunks of 32

// Async memory->LDS staging path (CDNA5): gate on builtin availability so the
// file always compiles; fallback is the sync global->VGPR->LDS path.
#if defined(__gfx1250__) &&                                                    \
    __has_builtin(__builtin_amdgcn_global_load_async_to_lds_b128) &&           \
    __has_builtin(__builtin_amdgcn_s_wait_asynccnt)
#define GRAM_ASYNC 1
#define NBUF 2
#else
#define NBUF 1
#endif

#define AS_G __attribute__((address_space(1)))
#define AS_L __attribute__((address_space(3)))

// ---------------------------------------------------------------------------
// Kernel 0: zero the scalar output
// ---------------------------------------------------------------------------
__global__ void init_out_kernel(float* out) {
    if (threadIdx.x == 0 && blockIdx.x == 0) out[0] = 0.0f;
}

// ---------------------------------------------------------------------------
// Kernel 1: row-normalize  Xn[i] = X[i] / max(||X[i]||, 1e-8)
// ---------------------------------------------------------------------------
__global__ __launch_bounds__(256) void norm_kernel(const float* __restrict__ X,
                                                   float* __restrict__ Xn) {
    const int i   = blockIdx.x;
    const int tid = threadIdx.x;
    const float* x = X + (size_t)i * DIM;

    float ss = 0.0f;
    #pragma unroll
    for (int j = tid; j < DIM; j += 256) { float v = x[j]; ss += v * v; }

    __shared__ float red[256];
    red[tid] = ss;
    __syncthreads();
    for (int o = 128; o > 0; o >>= 1) {
        if (tid < o) red[tid] += red[tid + o];
        __syncthreads();
    }
    const float inv = 1.0f / fmaxf(sqrtf(red[0]), 1e-8f);

    float* y = Xn + (size_t)i * DIM;
    #pragma unroll
    for (int j = tid; j < DIM; j += 256) y[j] = x[j] * inv;
}

// ---------------------------------------------------------------------------
// Kernel 2: G = Xn * Xn^T  (4096 x 4096 x 1024, FP32 via V_WMMA_F32_16X16X4_F32)
// block = 256 threads (8 waves) -> 128x128 output tile
// wave (wm in 0..3, wn in 0..1) -> 32x64 patch = 2x4 grid of 16x16 WMMA tiles
// K staged in LDS panels of 128x32 (stride 36 floats); async double-buffered
// via GLOBAL_LOAD_ASYNC_TO_LDS_B128 when available.
// ---------------------------------------------------------------------------
__global__ __launch_bounds__(256) void gram_kernel(const float* __restrict__ Xn,
                                                   float* __restrict__ G) {
    __shared__ float Ap[NBUF][128 * LSTRIDE];
    __shared__ float Bp[NBUF][128 * LSTRIDE];

    const int tid  = threadIdx.x;
    const int bm   = (blockIdx.x >> 5) * 128;   // 32x32 block grid
    const int bn   = (blockIdx.x & 31) * 128;
    const int wave = tid >> 5;
    const int lane = tid & 31;
    const int l15  = lane & 15;
    const int hl   = lane >> 4;                 // which 16-lane half
    const int wm   = wave >> 1;                 // 0..3
    const int wn   = wave & 1;                  // 0..1

    v8f acc[2][4];
    #pragma unroll
    for (int mi = 0; mi < 2; ++mi)
        #pragma unroll
        for (int ni = 0; ni < 4; ++ni)
            acc[mi][ni] = (v8f){0.f, 0.f, 0.f, 0.f, 0.f, 0.f, 0.f, 0.f};

#ifdef GRAM_ASYNC
    // 8 async-load instructions per wave per chunk (4 iters x 2 panels)
#define ISSUE_PANELS(K0, BUF)                                                  \
    {                                                                          \
        _Pragma("unroll")                                                      \
        for (int it_ = 0; it_ < 4; ++it_) {                                    \
            const int f_   = tid + it_ * 256;                                  \
            const int row_ = f_ >> 3;                                          \
            const int c4_  = (f_ & 7) << 2;                                    \
            __builtin_amdgcn_global_load_async_to_lds_b128(                    \
                (AS_G av4i*)&Xn[(size_t)(bm + row_) * DIM + (K0) + c4_],       \
                (AS_L av4i*)&Ap[BUF][row_ * LSTRIDE + c4_], 0, 0);             \
            __builtin_amdgcn_global_load_async_to_lds_b128(                    \
                (AS_G av4i*)&Xn[(size_t)(bn + row_) * DIM + (K0) + c4_],       \
                (AS_L av4i*)&Bp[BUF][row_ * LSTRIDE + c4_], 0, 0);             \
        }                                                                      \
    }

    ISSUE_PANELS(0, 0)
#endif

    for (int ci = 0; ci < NCHUNK; ++ci) {
#ifdef GRAM_ASYNC
        const int cur = ci & 1;
        if (ci + 1 < NCHUNK) {
            ISSUE_PANELS((ci + 1) * 32, cur ^ 1)        // prefetch next chunk
            __builtin_amdgcn_s_wait_asynccnt(8);        // retire previous 8
        } else {
            __builtin_amdgcn_s_wait_asynccnt(0);
        }
#else
        const int cur = 0;
        const int k0  = ci * 32;
        #pragma unroll
        for (int it = 0; it < 4; ++it) {
            const int f   = tid + it * 256;             // 0..1023 float4 slots
            const int row = f >> 3;
            const int c4  = (f & 7) << 2;
            const float4 va = *(const float4*)&Xn[(size_t)(bm + row) * DIM + k0 + c4];
            const float4 vb = *(const float4*)&Xn[(size_t)(bn + row) * DIM + k0 + c4];
            *(float4*)&Ap[0][row * LSTRIDE + c4] = va;
            *(float4*)&Bp[0][row * LSTRIDE + c4] = vb;
        }
#endif
        __syncthreads();

        #pragma unroll
        for (int kk = 0; kk < 8; ++kk) {
            const int col = (kk << 2) + (hl << 1);      // K = 4*kk + 2*half + {0,1}
            v2f a[2], b[4];
            #pragma unroll
            for (int mi = 0; mi < 2; ++mi)
                a[mi] = *(const v2f*)&Ap[cur][(wm * 32 + mi * 16 + l15) * LSTRIDE + col];
            #pragma unroll
            for (int ni = 0; ni < 4; ++ni)
                b[ni] = *(const v2f*)&Bp[cur][(wn * 64 + ni * 16 + l15) * LSTRIDE + col];
            #pragma unroll
            for (int mi = 0; mi < 2; ++mi)
                #pragma unroll
                for (int ni = 0; ni < 4; ++ni)
                    acc[mi][ni] = __builtin_amdgcn_wmma_f32_16x16x4_f32(
                        false, a[mi], false, b[ni], (short)0, acc[mi][ni], false, false);
        }
        __syncthreads();
    }

    // store: C/D layout -> VGPR v holds M = v + 8*half, N = lane&15
    #pragma unroll
    for (int mi = 0; mi < 2; ++mi)
        #pragma unroll
        for (int ni = 0; ni < 4; ++ni)
            #pragma unroll
            for (int v = 0; v < 8; ++v) {
                const int gr = bm + wm * 32 + mi * 16 + v + hl * 8;
                const int gc = bn + wn * 64 + ni * 16 + l15;
                G[(size_t)gr * N_ROWS + gc] = acc[mi][ni][v];
            }
}

// ---------------------------------------------------------------------------
// threefry2x32-20 with key = jax.random.key(42) -> (k0,k1) = (0,42)
// ---------------------------------------------------------------------------
__device__ __forceinline__ unsigned tf_rotl(unsigned v, int d) {
    return (v << d) | (v >> (32 - d));
}

__device__ __forceinline__ unsigned jax_bits(unsigned pos) {
    const unsigned HALF = 7534592u;   // (4096 * 3679) / 2
    const unsigned k0 = 0u, k1 = 42u;
    const unsigned k2 = 0x1BD11BDAu ^ k0 ^ k1;
    unsigned c0, c1;
    const bool lo = pos < HALF;
    if (lo) { c0 = pos;        c1 = pos + HALF; }
    else    { c0 = pos - HALF; c1 = pos;        }
    unsigned x0 = c0 + k0, x1 = c1 + k1;
#define TFR(r) { x0 += x1; x1 = tf_rotl(x1, r); x1 ^= x0; }
    TFR(13) TFR(15) TFR(26) TFR(6)   x0 += k1; x1 += k2 + 1u;
    TFR(17) TFR(29) TFR(16) TFR(24)  x0 += k2; x1 += k0 + 2u;
    TFR(13) TFR(15) TFR(26) TFR(6)   x0 += k0; x1 += k1 + 3u;
    TFR(17) TFR(29) TFR(16) TFR(24)  x0 += k1; x1 += k2 + 4u;
    TFR(13) TFR(15) TFR(26) TFR(6)   x0 += k2; x1 += k0 + 5u;
#undef TFR
    return lo ? x0 : x1;
}

// ---------------------------------------------------------------------------
// Kernel 3: fused triplet loss. One block per row i.
// per = max(G[i,ncol] - G[i,pcol] + margin, 0), summed, rows with i%8==7 skipped.
// ---------------------------------------------------------------------------
__global__ __launch_bounds__(256) void loss_kernel(const float* __restrict__ G,
                                                   float* __restrict__ out) {
    const int i   = blockIdx.x;
    const int tid = threadIdx.x;
    const int p   = 7 - (i & 7);          // positive count for this row
    if (p == 0) return;                   // invalid row (no positives)
    const int   c   = i >> 3;             // class id
    const int   L   = c, R = (NCLASS - 1) - c;
    const int   S   = (L > R) ? (L - R) : (R - L);
    const int   mlr = (L < R) ? L : R;
    const float pf  = (float)p;
    const unsigned base = (unsigned)i * (unsigned)N_NEGS;
    const float* g = G + (size_t)i * N_ROWS;

    float acc = 0.0f;
    for (int s = tid; s < N_NEGS; s += 256) {
        // ---- mined negative column (closed-form argsort of |dt| desc, stable) ----
        const int b = s >> 3;
        int cls;
        if (b < S) {
            cls = (L > R) ? b : ((NCLASS - 1) - b);
        } else {
            const int e = b - S;
            const int d = mlr - (e >> 1);
            cls = (e & 1) ? (c + d) : (c - d);
        }
        const int ncol = (cls << 3) + (s & 7);

        // ---- resampled positive column via threefry2x32 uniform ----
        const unsigned bits = jax_bits(base + (unsigned)s);
        const float u = __uint_as_float((bits >> 9) | 0x3f800000u) - 1.0f;
        int ridx = (int)(u * pf);
        ridx = (ridx > p - 1) ? (p - 1) : ridx;
        const int pcol = i + 1 + ridx;

        // ---- hinge: d_pos - d_neg + m = (1-Gp) - (1-Gn) + m ----
        const float t = g[ncol] - g[pcol] + MARGIN;
        acc += (t > 0.0f) ? t : 0.0f;
    }

    __shared__ float red[256];
    red[tid] = acc;
    __syncthreads();
    for (int o = 128; o > 0; o >>= 1) {
        if (tid < o) red[tid] += red[tid + o];
        __syncthreads();
    }
    if (tid == 0) atomicAdd(out, red[0]);
}

// ---------------------------------------------------------------------------
// launch
// ---------------------------------------------------------------------------
extern "C" void kernel_launch(void* const* d_in, const int* in_sizes, int n_in,
                              void* d_out, int out_size, void* d_ws, size_t ws_size,
                              hipStream_t stream) {
    const float* X   = (const float*)d_in[0];          // [4096,1024] f32
    float*       out = (float*)d_out;                  // scalar f32

    // workspace: [0,16MB) normalized rows, [16MB,80MB) Gram matrix
    float* Xn = (float*)d_ws;
    float* G  = (float*)((char*)d_ws + (size_t)N_ROWS * DIM * sizeof(float));

    init_out_kernel<<<1, 64, 0, stream>>>(out);
    norm_kernel<<<N_ROWS, 256, 0, stream>>>(X, Xn);
    gram_kernel<<<(N_ROWS / 128) * (N_ROWS / 128), 256, 0, stream>>>(Xn, G);
    loss_kernel<<<N_ROWS, 256, 0, stream>>>(G, out);
}